// hex_pooling_1949915152424
// MI455X (gfx1250) — compile-verified
//
#include <hip/hip_runtime.h>

// hex pooling: out[r, :] = max_k x[hex_idx[r, k], :],  r in [0, L), L = (N+6)//4
// N = 40962, K = 7, C = 512, L = 10242. Pure gather + 7-way max: memory-bound,
// x (84 MB) fits in the MI455X 192 MB L2 -> HBM floor ~105 MB @ 23.3 TB/s.

#define KNB 7
#define CH 512
#define TPB 128           // one float4 (16B) per lane -> 512 floats per row
#define ROWS_PER_BLOCK 2  // row B is prefetched while row A computes

typedef float v4f __attribute__((ext_vector_type(4)));

__global__ __launch_bounds__(TPB) void hex_pool_max_kernel(
    const float* __restrict__ x,
    const long long* __restrict__ hex_idx,
    float* __restrict__ out,
    int L)
{
    const int tid  = threadIdx.x;
    const int c4   = tid * 4;                      // float column offset
    const int rowA = blockIdx.x * ROWS_PER_BLOCK;
    if (rowA >= L) return;
    const int rowB   = rowA + 1;
    const bool hasB  = rowB < L;                   // uniform per block

    // ---- indices: uniform addresses -> scalar (s_load) path ----
    long long ia[KNB], ib[KNB];
    const long long* __restrict__ ha = hex_idx + (long long)rowA * KNB;
    const long long* __restrict__ hb = hex_idx + (long long)(hasB ? rowB : rowA) * KNB;
#pragma unroll
    for (int k = 0; k < KNB; ++k) ia[k] = ha[k];
#pragma unroll
    for (int k = 0; k < KNB; ++k) ib[k] = hb[k];

    // ---- gfx1250 prefetch: pull row B's gather targets toward L2/L0 while
    //      row A's loads + max tree run (emits global_prefetch_b8) ----
#pragma unroll
    for (int k = 0; k < KNB; ++k)
        __builtin_prefetch(x + ib[k] * (long long)CH + c4, 0, 3);

    // ---- row A: 7 independent global_load_b128 in flight, then max tree ----
    {
        v4f v0 = *(const v4f*)(x + ia[0] * (long long)CH + c4);
        v4f v1 = *(const v4f*)(x + ia[1] * (long long)CH + c4);
        v4f v2 = *(const v4f*)(x + ia[2] * (long long)CH + c4);
        v4f v3 = *(const v4f*)(x + ia[3] * (long long)CH + c4);
        v4f v4 = *(const v4f*)(x + ia[4] * (long long)CH + c4);
        v4f v5 = *(const v4f*)(x + ia[5] * (long long)CH + c4);
        v4f v6 = *(const v4f*)(x + ia[6] * (long long)CH + c4);
        v4f m01 = __builtin_elementwise_max(v0, v1);
        v4f m23 = __builtin_elementwise_max(v2, v3);
        v4f m45 = __builtin_elementwise_max(v4, v5);
        v4f m   = __builtin_elementwise_max(
                      __builtin_elementwise_max(m01, m23),
                      __builtin_elementwise_max(m45, v6));
        // streaming output: non-temporal store keeps x resident in L2
        __builtin_nontemporal_store(m, (v4f*)(out + (long long)rowA * CH + c4));
    }

    // ---- row B (already prefetched) ----
    if (hasB) {
        v4f v0 = *(const v4f*)(x + ib[0] * (long long)CH + c4);
        v4f v1 = *(const v4f*)(x + ib[1] * (long long)CH + c4);
        v4f v2 = *(const v4f*)(x + ib[2] * (long long)CH + c4);
        v4f v3 = *(const v4f*)(x + ib[3] * (long long)CH + c4);
        v4f v4 = *(const v4f*)(x + ib[4] * (long long)CH + c4);
        v4f v5 = *(const v4f*)(x + ib[5] * (long long)CH + c4);
        v4f v6 = *(const v4f*)(x + ib[6] * (long long)CH + c4);
        v4f m01 = __builtin_elementwise_max(v0, v1);
        v4f m23 = __builtin_elementwise_max(v2, v3);
        v4f m45 = __builtin_elementwise_max(v4, v5);
        v4f m   = __builtin_elementwise_max(
                      __builtin_elementwise_max(m01, m23),
                      __builtin_elementwise_max(m45, v6));
        __builtin_nontemporal_store(m, (v4f*)(out + (long long)rowB * CH + c4));
    }
}

extern "C" void kernel_launch(void* const* d_in, const int* in_sizes, int n_in,
                              void* d_out, int out_size, void* d_ws, size_t ws_size,
                              hipStream_t stream)
{
    const float*     x   = (const float*)d_in[0];
    const long long* idx = (const long long*)d_in[1];  // int64 per reference
    float*           out = (float*)d_out;

    // derive L from the index array: n = rows of hex_idx, L = (n+6)//4
    const long long n = (long long)in_sizes[1] / KNB;
    const int L = (int)((n + 6) / 4);                  // 10242 for n=40962

    dim3 grid((L + ROWS_PER_BLOCK - 1) / ROWS_PER_BLOCK);
    dim3 block(TPB);
    hipLaunchKernelGGL(hex_pool_max_kernel, grid, block, 0, stream, x, idx, out, L);
}